// GNNModel_30382598652232
// MI455X (gfx1250) — compile-verified
//
#include <hip/hip_runtime.h>

#define N_NODES 100000
#define E_EDGES 1600000
#define F_DIM   128

typedef __attribute__((ext_vector_type(2))) float v2f;
typedef __attribute__((ext_vector_type(8))) float v8f;

// ---------------- graph-norm precompute ----------------

__global__ void deg_init_kernel(float* __restrict__ deg) {
    int n = blockIdx.x * 256 + threadIdx.x;
    if (n < N_NODES) deg[n] = 1.0f;   // self loop
}

__global__ void deg_count_kernel(const int* __restrict__ dst, float* __restrict__ deg) {
    int e = blockIdx.x * 256 + threadIdx.x;
    if (e < E_EDGES) atomicAdd(&deg[dst[e]], 1.0f);
}

__global__ void dis_kernel(float* __restrict__ deg) {
    int n = blockIdx.x * 256 + threadIdx.x;
    if (n < N_NODES) deg[n] = rsqrtf(deg[n]);   // deg >= 1 always (self loops)
}

__global__ void norm_kernel(const int* __restrict__ src, const int* __restrict__ dst,
                            const float* __restrict__ dis, float* __restrict__ norm) {
    int e = blockIdx.x * 256 + threadIdx.x;
    if (e < E_EDGES) norm[e] = dis[src[e]] * dis[dst[e]];
}

// ---------------- dense GEMM: Y[N,128] = act(X)[N,128] @ W[128,128] ----------------
// One wave computes a 16x128 output tile: 8 n-tiles, K-loop of 32 x (16x16x4) f32 WMMA.
// W is staged in LDS (64 KB) in a K-pair interleaved layout:
//   wlds[(k>>1)*256 + n*2 + 0] = W[k][n], wlds[(k>>1)*256 + n*2 + 1] = W[k+1][n]
// so each B fragment (rows kb, kb+1 of column n) is a single aligned ds_load_b64
// landing in an adjacent even VGPR pair (no re-pairing movs before the WMMA).

template <bool APPLY_RELU>
__global__ __launch_bounds__(256) void gemm_wmma_kernel(const float* __restrict__ X,
                                                        const float* __restrict__ W,
                                                        float* __restrict__ Y) {
    extern __shared__ float wlds[];   // 64 pairs x 128 cols x 2 = 16384 floats

    const int t = threadIdx.x;
    for (int i = t; i < 64 * F_DIM; i += 256) {   // i = k2*128 + n
        const int k2 = i >> 7;
        const int n  = i & 127;
        v2f w;
        w.x = W[(2 * k2) * F_DIM + n];
        w.y = W[(2 * k2 + 1) * F_DIM + n];
        *(v2f*)&wlds[(size_t)i * 2] = w;
    }
    __syncthreads();

    const int wave  = t >> 5;
    const int lane  = t & 31;
    const int l15   = lane & 15;     // lane within half-wave
    const int lhalf = lane >> 4;     // 0 or 1
    const int rowbase = (blockIdx.x * 8 + wave) * 16;
    if (rowbase >= N_NODES) return;  // wave-uniform guard (EXEC stays all-1s for WMMA)

    v8f zero = {};
    v8f acc[8];
#pragma unroll
    for (int nt = 0; nt < 8; ++nt) acc[nt] = zero;

    // A fragment (32-bit A 16x4 layout): lane L holds row M = L&15;
    // VGPR0/1 hold K = kb, kb+1 with kb = 4*ks + 2*(L>>4)
    const float* xrow = X + (size_t)(rowbase + l15) * F_DIM + 2 * lhalf;

    // B fragment base: element index (p*128 + n)*2 with p = 2*ks + lhalf,
    // n = nt*16 + l15  ->  lane-dependent part = 256*lhalf + 2*l15 floats;
    // (ks,nt)-dependent part = 512*ks + 32*nt floats (folds into DS immediate).
    const float* bbase = wlds + 256 * lhalf + 2 * l15;

#pragma unroll 4
    for (int ks = 0; ks < 32; ++ks) {
        v2f a = *(const v2f*)(xrow + 4 * ks);   // X[row][kb], X[row][kb+1] (8B aligned)
        if (APPLY_RELU) {
            a.x = fmaxf(a.x, 0.0f);
            a.y = fmaxf(a.y, 0.0f);
        }
#pragma unroll
        for (int nt = 0; nt < 8; ++nt) {
            v2f b = *(const v2f*)(bbase + 512 * ks + 32 * nt);   // ds_load_b64
            acc[nt] = __builtin_amdgcn_wmma_f32_16x16x4_f32(
                false, a, false, b, (short)0, acc[nt], false, false);
        }
    }

    // C/D 16x16 f32 layout: VGPR v -> M = v + 8*(lane>>4), N = (lane&15) (+ 16*nt)
#pragma unroll
    for (int nt = 0; nt < 8; ++nt) {
#pragma unroll
        for (int v = 0; v < 8; ++v) {
            const int m = v + 8 * lhalf;
            Y[(size_t)(rowbase + m) * F_DIM + nt * 16 + l15] = acc[nt][v];
        }
    }
}

// ---------------- self-loop seed + bias: y = xw * dis^2 + b ----------------

__global__ void init_out_kernel(const float* __restrict__ xw, const float* __restrict__ dis,
                                const float* __restrict__ bias, float* __restrict__ y) {
    unsigned i = blockIdx.x * 256u + threadIdx.x;
    if (i < (unsigned)N_NODES * F_DIM) {
        unsigned n = i >> 7, f = i & 127u;
        float d = dis[n];
        y[i] = xw[i] * d * d + bias[f];
    }
}

// ---------------- edge scatter: y[dst] += xw[src] * norm[e], 128 threads/edge ----------------

__global__ void scatter_kernel(const int* __restrict__ src, const int* __restrict__ dst,
                               const float* __restrict__ norm, const float* __restrict__ xw,
                               float* __restrict__ y) {
    unsigned i = blockIdx.x * 256u + threadIdx.x;   // < E*128 = 204.8M < 2^31
    unsigned e = i >> 7, f = i & 127u;
    if (e < E_EDGES) {
        unsigned s = (unsigned)src[e] * 128u + f;
        unsigned d = (unsigned)dst[e] * 128u + f;
        atomicAdd(&y[d], xw[s] * norm[e]);
    }
}

// ---------------- launch ----------------

extern "C" void kernel_launch(void* const* d_in, const int* in_sizes, int n_in,
                              void* d_out, int out_size, void* d_ws, size_t ws_size,
                              hipStream_t stream) {
    const int*   edge = (const int*)d_in[0];       // [2, E] int32
    const int*   src  = edge;
    const int*   dst  = edge + E_EDGES;
    // d_in[1] = edge_attr (unused by reference)
    const float* emb  = (const float*)d_in[2];     // [N, 128]
    const float* W1   = (const float*)d_in[3];     // [128, 128]
    const float* b1   = (const float*)d_in[4];     // [128]
    const float* W2   = (const float*)d_in[5];     // [128, 128]
    const float* b2   = (const float*)d_in[6];     // [128]
    float*       out  = (float*)d_out;             // [N, 128]

    float* dis  = (float*)d_ws;                    // N floats (deg -> rsqrt in place)
    float* norm = dis  + N_NODES;                  // E floats
    float* xw   = norm + E_EDGES;                  // N*128 floats
    float* y1   = xw   + (size_t)N_NODES * F_DIM;  // N*128 floats

    const int TB = 256;
    const int nodeBlocks = (N_NODES + TB - 1) / TB;
    const int edgeBlocks = (E_EDGES + TB - 1) / TB;
    const int nfBlocks   = (N_NODES * F_DIM + TB - 1) / TB;       // 50000
    const int efBlocks   = (int)(((size_t)E_EDGES * F_DIM + TB - 1) / TB); // 800000
    const int gemmBlocks = (N_NODES / 16 + 7) / 8;                // 782
    const size_t ldsBytes = (size_t)F_DIM * F_DIM * sizeof(float); // 64 KB

    // graph normalization (shared by both layers)
    deg_init_kernel <<<nodeBlocks, TB, 0, stream>>>(dis);
    deg_count_kernel<<<edgeBlocks, TB, 0, stream>>>(dst, dis);
    dis_kernel      <<<nodeBlocks, TB, 0, stream>>>(dis);
    norm_kernel     <<<edgeBlocks, TB, 0, stream>>>(src, dst, dis, norm);

    // layer 1: xw = emb @ W1 ; y1 = scatter(xw) + self + b1
    gemm_wmma_kernel<false><<<gemmBlocks, TB, ldsBytes, stream>>>(emb, W1, xw);
    init_out_kernel <<<nfBlocks, TB, 0, stream>>>(xw, dis, b1, y1);
    scatter_kernel  <<<efBlocks, TB, 0, stream>>>(src, dst, norm, xw, y1);

    // layer 2: xw = relu(y1) @ W2 ; out = scatter(xw) + self + b2
    gemm_wmma_kernel<true><<<gemmBlocks, TB, ldsBytes, stream>>>(y1, W2, xw);
    init_out_kernel <<<nfBlocks, TB, 0, stream>>>(xw, dis, b2, out);
    scatter_kernel  <<<efBlocks, TB, 0, stream>>>(src, dst, norm, xw, out);
}